// DNDT_85169201479696
// MI455X (gfx1250) — compile-verified
//
#include <hip/hip_runtime.h>

typedef float v2f __attribute__((ext_vector_type(2)));
typedef float v8f __attribute__((ext_vector_type(8)));

#define NFEAT   6
#define NCLASS  10
#define TILE_M  16
#define WAVES_PER_BLOCK 8

// One wave (32 lanes) computes a 16-row batch tile:
//   out[16x10] = leaf[16x4096] @ leaf_score[4096x10]
// leaf is never materialized: each lane holds the 6x4 softmax bins of its row
// in registers and generates A-fragments via a base-4 digit prefix product.
__global__ __launch_bounds__(WAVES_PER_BLOCK * 32)
void dndt_wmma_kernel(const float* __restrict__ x,      // [B,6]
                      const float* __restrict__ cuts,   // [6,3]
                      const float* __restrict__ ls,     // [4096,10]
                      const float* __restrict__ temp,   // [1]
                      float* __restrict__ out,          // [B,10]
                      int batch)
{
    const int lane   = threadIdx.x & 31;
    const int wave   = threadIdx.x >> 5;
    const int tile   = blockIdx.x * WAVES_PER_BLOCK + wave;  // 16-row tile
    if (tile * TILE_M >= batch) return;                      // uniform per wave

    const int  ncol  = lane & 15;     // A: row M / C,D: column N
    const int  khalf = lane >> 4;     // 0: K={0,1}, 1: K={2,3} within a chunk
    const int  row   = tile * TILE_M + ncol;

    const float invT = 1.0f / temp[0];

    // ---- per-row softmax bins, all in registers (lanes L and L+16 redundantly
    // compute the same row; reads coalesce/broadcast through WGP$) ----
    float bin[NFEAT][4];
    #pragma unroll
    for (int f = 0; f < NFEAT; ++f) {
        float c0 = cuts[f * 3 + 0];
        float c1 = cuts[f * 3 + 1];
        float c2 = cuts[f * 3 + 2];
        float lo  = fminf(fminf(c0, c1), c2);
        float hi  = fmaxf(fmaxf(c0, c1), c2);
        float mid = (c0 + c1 + c2) - lo - hi;

        float xv = x[row * NFEAT + f];
        // h_d = (x*(d+1) + cumsum(-sorted_cuts))/T
        float h0 = (xv * 1.0f) * invT;
        float h1 = (xv * 2.0f - lo) * invT;
        float h2 = (xv * 3.0f - lo - mid) * invT;
        float h3 = (xv * 4.0f - lo - mid - hi) * invT;

        float mx = fmaxf(fmaxf(h0, h1), fmaxf(h2, h3));
        const float L2E = 1.4426950408889634f;
        float e0 = exp2f((h0 - mx) * L2E);
        float e1 = exp2f((h1 - mx) * L2E);
        float e2 = exp2f((h2 - mx) * L2E);
        float e3 = exp2f((h3 - mx) * L2E);
        float inv = 1.0f / (e0 + e1 + e2 + e3);
        bin[f][0] = e0 * inv;
        bin[f][1] = e1 * inv;
        bin[f][2] = e2 * inv;
        bin[f][3] = e3 * inv;
    }

    // B operand base pointer: column clamped to 9 (cols 10..15 of the padded
    // 16-wide B duplicate col 9; they only feed C columns we never store, and
    // the clamp keeps every address inside leaf_score).
    const int   bcol = (ncol < NCLASS) ? ncol : (NCLASS - 1);
    const float* __restrict__ bp = ls + bcol + khalf * 2 * NCLASS;

    const float b5a = bin[5][2 * khalf + 0];   // K-lane d5 digits
    const float b5b = bin[5][2 * khalf + 1];

    v8f acc = {};   // 16x16 f32 accumulator

    // leaf index j = ((((d0*4+d1)*4+d2)*4+d3)*4+d4)*4 + d5 ; d5 == WMMA K dim
    for (int d0 = 0; d0 < 4; ++d0) {
        float p0 = bin[0][d0];
        for (int d1 = 0; d1 < 4; ++d1) {
            float p1 = p0 * bin[1][d1];
            for (int d2 = 0; d2 < 4; ++d2) {
                float p2 = p1 * bin[2][d2];
                #pragma unroll
                for (int d3 = 0; d3 < 4; ++d3) {
                    float p3 = p2 * bin[3][d3];
                    #pragma unroll
                    for (int d4 = 0; d4 < 4; ++d4) {
                        float p4 = p3 * bin[4][d4];
                        int kbase = ((((d0 * 4 + d1) * 4 + d2) * 4 + d3) * 4 + d4) * 4;

                        v2f A = { p4 * b5a, p4 * b5b };

                        const float* bpk = bp + kbase * NCLASS;
                        v2f B = { bpk[0], bpk[NCLASS] };

                        acc = __builtin_amdgcn_wmma_f32_16x16x4_f32(
                                  false, A, false, B,
                                  (short)0, acc, false, false);
                    }
                }
            }
        }
    }

    // D layout: lane n = lane&15 ; VGPR v -> row m = (lane>>4)*8 + v
    if (ncol < NCLASS) {
        #pragma unroll
        for (int v = 0; v < 8; ++v) {
            int m = khalf * 8 + v;
            out[(tile * TILE_M + m) * NCLASS + ncol] = acc[v];
        }
    }
}

extern "C" void kernel_launch(void* const* d_in, const int* in_sizes, int n_in,
                              void* d_out, int out_size, void* d_ws, size_t ws_size,
                              hipStream_t stream) {
    const float* x    = (const float*)d_in[0];   // [B,6]
    const float* cuts = (const float*)d_in[1];   // [6,3]
    const float* ls   = (const float*)d_in[2];   // [4096,10]
    const float* temp = (const float*)d_in[3];   // [1]
    float*       out  = (float*)d_out;           // [B,10]

    const int batch = in_sizes[0] / NFEAT;                     // 16384
    const int tiles = (batch + TILE_M - 1) / TILE_M;           // 1024
    const int grid  = (tiles + WAVES_PER_BLOCK - 1) / WAVES_PER_BLOCK;  // 128

    dndt_wmma_kernel<<<grid, WAVES_PER_BLOCK * 32, 0, stream>>>(
        x, cuts, ls, temp, out, batch);
}